// Discriminator_29729763623023
// MI455X (gfx1250) — compile-verified
//
#include <hip/hip_runtime.h>

// ---------------------------------------------------------------------------
// LSTM scan (B=2048, T=1024, I=4, H=16) + per-step dense sigmoid head.
// One wave32 per 16-row batch tile; V_WMMA_F32_16X16X4_F32 for all matmuls.
// Transposed mapping: M = gate columns (weights in A, loop-invariant),
// N = batch (data in B). The h-state recycles into the next step's B operand
// with v_permlanex16 half-wave swaps -- no LDS round-trip in the recurrence.
// Sigmoid gates use sigmoid(x) = 0.5*tanh(x/2)+0.5 with the x/2 folded into
// pre-scaled weights/biases (exact), so each gate costs one TRANS op.
// ---------------------------------------------------------------------------

#define B_   2048
#define T_   1024
#define I_   4
#define H_   16
#define TCH  32            // timesteps staged per LDS chunk
#define NTILE (B_ / 16)    // 128 waves

typedef float v2f __attribute__((ext_vector_type(2)));
typedef float v8f __attribute__((ext_vector_type(8)));

// LDS row strides (bytes), padded against bank conflicts, 16B-aligned.
#define XROW 528           // 32 t * 16B + 16B pad   (b128-aligned)
#define OROW 144           // 32 f32 + 16B pad       (b128-aligned)

__device__ __forceinline__ float tanh_fast(float x) {
#if __has_builtin(__builtin_amdgcn_tanhf)
  return __builtin_amdgcn_tanhf(x);
#else
  float r;
  asm volatile("v_tanh_f32 %0, %1" : "=v"(r) : "v"(x));
  return r;
#endif
}

// sigmoid(2y) = 0.5*tanh(y) + 0.5 ; caller supplies y = x/2 via pre-scaled W,b
__device__ __forceinline__ float sig_from_half(float y) {
  return __builtin_fmaf(tanh_fast(y), 0.5f, 0.5f);
}

// lane i <-> lane i^16 exchange (identity selects into the opposite row)
__device__ __forceinline__ float swap_half(float v) {
#if __has_builtin(__builtin_amdgcn_permlanex16)
  int r = __builtin_amdgcn_permlanex16(
      __builtin_bit_cast(int, v), __builtin_bit_cast(int, v),
      0x76543210, 0xfedcba98, false, false);
  return __builtin_bit_cast(float, r);
#else
  return __shfl_xor(v, 16, 32);
#endif
}

__device__ __forceinline__ v8f splat8(float v) {
  v8f r;
#pragma unroll
  for (int i = 0; i < 8; ++i) r[i] = v;
  return r;
}

__global__ __launch_bounds__(32)
void lstm_scan_kernel(const float* __restrict__ x,
                      const float* __restrict__ W_ih,
                      const float* __restrict__ W_hh,
                      const float* __restrict__ b_ih,
                      const float* __restrict__ b_hh,
                      const float* __restrict__ W_d,
                      const float* __restrict__ b_d,
                      float* __restrict__ out) {
  __shared__ __align__(16) unsigned char lds[16 * XROW + 16 * OROW];
  unsigned char* xstage = lds;                 // x tile [b][tt][i]
  unsigned char* ostage = lds + 16 * XROW;     // out tile [b][tt]

  const int lane = threadIdx.x;   // 0..31
  const int n    = lane & 15;     // N index: batch row within tile; also M row
  const int q    = lane >> 4;     // lane half
  const int kh   = q * 2;         // per-half K base (f32 16x16x4 A/B layout)
  const bool hi  = (q != 0);
  const int b0   = blockIdx.x * 16;

  // ---- loop-invariant A operands (weights), pre-scaled ----
  // Gate blocks: 0=i, 1=f, 2=g(tanh, unscaled), 3=o. Sigmoid blocks and the
  // dense head are scaled by 0.5 so the WMMA emits x/2 directly.
  // A layout (16x4 f32): lane holds A[m = lane%16][kh + {0,1}], kh = 2*q.
  v2f Ah[4][4];   // [gate block][K chunk]
  v2f Axw[4];     // input projection A = W_ih block (K = I = 4, one chunk)
  v2f Ad[4];      // dense head A = 0.5*W_d broadcast over rows
  v8f bbv[4];     // bias per gate element j = r + 8q
#pragma unroll
  for (int g = 0; g < 4; ++g) {
    const float sc = (g == 2) ? 1.0f : 0.5f;
    const float* wi = W_ih + (g * 16 + n) * I_ + kh;
    Axw[g][0] = sc * wi[0];
    Axw[g][1] = sc * wi[1];
#pragma unroll
    for (int c = 0; c < 4; ++c) {
      const float* wh = W_hh + (g * 16 + n) * H_ + c * 4 + kh;
      Ah[g][c][0] = sc * wh[0];
      Ah[g][c][1] = sc * wh[1];
    }
#pragma unroll
    for (int r = 0; r < 8; ++r)
      bbv[g][r] =
          sc * (b_ih[g * 16 + r + 8 * q] + b_hh[g * 16 + r + 8 * q]);
  }
#pragma unroll
  for (int c = 0; c < 4; ++c) {
    Ad[c][0] = 0.5f * W_d[c * 4 + kh];
    Ad[c][1] = 0.5f * W_d[c * 4 + kh + 1];
  }
  const float bdh = 0.5f * b_d[0];

  v8f cst = splat8(0.f);  // cell state, layout [j = r + 8q][b = lane%16]
  v2f bhc[4];             // B chunks of h_{t-1}: B[k = c*4+kh+{0,1}][n = b]
#pragma unroll
  for (int c = 0; c < 4; ++c) { bhc[c][0] = 0.f; bhc[c][1] = 0.f; }

  for (int t0 = 0; t0 < T_; t0 += TCH) {
    // ---- stage x[b0+j][t0..t0+31][0..3] into LDS (coalesced b128) ----
#pragma unroll
    for (int j = 0; j < 16; ++j) {
      const float4* gsrc =
          (const float4*)(x + (((size_t)(b0 + j)) * T_ + t0) * I_);
      float4 v = gsrc[lane];  // one float4 == one timestep's 4 inputs
      *(float4*)(xstage + j * XROW + lane * 16) = v;
    }
    if (t0 + TCH < T_) {
      // one global_prefetch covers next chunk (16 rows x 512B, 256B/lane)
      __builtin_prefetch(
          x + (((size_t)(b0 + (lane >> 1))) * T_ + (t0 + TCH)) * I_ +
              (lane & 1) * 64,
          0, 0);
    }

    for (int tt = 0; tt < TCH; ++tt) {
      // x B operand: B[k = kh + {0,1}][n = b]  (hidden behind recurrent WMMAs)
      v2f bx = *(const v2f*)(xstage + n * XROW + (tt * I_ + kh) * 4);

      // gates^T = bias + W_hh @ h^T + W_ih @ x^T ; bias rides in as C of the
      // first WMMA of each chain (no accumulator-init moves).
      v8f acc[4];
#pragma unroll
      for (int g = 0; g < 4; ++g)
        acc[g] = __builtin_amdgcn_wmma_f32_16x16x4_f32(
            false, Ah[g][0], false, bhc[0], (short)0, bbv[g], false, false);
#pragma unroll
      for (int c = 1; c < 4; ++c)
#pragma unroll
        for (int g = 0; g < 4; ++g)
          acc[g] = __builtin_amdgcn_wmma_f32_16x16x4_f32(
              false, Ah[g][c], false, bhc[c], (short)0, acc[g], false, false);
#pragma unroll
      for (int g = 0; g < 4; ++g)
        acc[g] = __builtin_amdgcn_wmma_f32_16x16x4_f32(
            false, Axw[g], false, bx, (short)0, acc[g], false, false);

      // elementwise gate math; element (j = r + 8q, b = lane%16)
      v8f hv;
#pragma unroll
      for (int r = 0; r < 8; ++r) {
        float ig = sig_from_half(acc[0][r]);
        float fg = sig_from_half(acc[1][r]);
        float gg = tanh_fast(acc[2][r]);
        float og = sig_from_half(acc[3][r]);
        float cv = __builtin_fmaf(fg, cst[r], ig * gg);
        cst[r] = cv;
        hv[r] = og * tanh_fast(cv);
      }

      // Rebuild B chunks of the NEW h with half-wave swaps (pure VALU).
      // lane holds h[b][j = r + 8q]; chunk c needs h[b][4c + 2q + {0,1}].
      float s0 = swap_half(hv[0]), s1 = swap_half(hv[1]);
      float s2 = swap_half(hv[2]), s3 = swap_half(hv[3]);
      float s4 = swap_half(hv[4]), s5 = swap_half(hv[5]);
      float s6 = swap_half(hv[6]), s7 = swap_half(hv[7]);
      bhc[0][0] = hi ? s2 : hv[0];
      bhc[0][1] = hi ? s3 : hv[1];
      bhc[1][0] = hi ? s6 : hv[4];
      bhc[1][1] = hi ? s7 : hv[5];
      bhc[2][0] = hi ? hv[2] : s0;
      bhc[2][1] = hi ? hv[3] : s1;
      bhc[3][0] = hi ? hv[6] : s4;
      bhc[3][1] = hi ? hv[7] : s5;

      // dense head: out^T[.][b] = 0.5*W_d @ h_t^T ; A rows replicated, so
      // every lane's vgpr0 holds out[b]/2 -- both half-waves store the same
      // value to the same address (deterministic), no exec-mask dance.
      v8f d = splat8(0.f);
#pragma unroll
      for (int c = 0; c < 4; ++c)
        d = __builtin_amdgcn_wmma_f32_16x16x4_f32(
            false, Ad[c], false, bhc[c], (short)0, d, false, false);

      float s = sig_from_half(d[0] + bdh);
      *(float*)(ostage + n * OROW + tt * 4) = s;
    }

    // ---- flush out tile [16 b][32 t] -> global, coalesced b128 ----
#pragma unroll
    for (int j = 0; j < 4; ++j) {
      int idx = j * 32 + lane;  // 16B units, 0..127
      int m   = idx >> 3;
      int off = idx & 7;
      float4 v = *(const float4*)(ostage + m * OROW + off * 16);
      *(float4*)(out + ((size_t)(b0 + m)) * T_ + t0 + off * 4) = v;
    }
  }
}

extern "C" void kernel_launch(void* const* d_in, const int* in_sizes, int n_in,
                              void* d_out, int out_size, void* d_ws,
                              size_t ws_size, hipStream_t stream) {
  (void)in_sizes; (void)n_in; (void)d_ws; (void)ws_size; (void)out_size;
  const float* x    = (const float*)d_in[0];
  const float* W_ih = (const float*)d_in[1];
  const float* W_hh = (const float*)d_in[2];
  const float* b_ih = (const float*)d_in[3];
  const float* b_hh = (const float*)d_in[4];
  const float* W_d  = (const float*)d_in[5];
  const float* b_d  = (const float*)d_in[6];
  float* out = (float*)d_out;

  lstm_scan_kernel<<<NTILE, 32, 0, stream>>>(x, W_ih, W_hh, b_ih, b_hh, W_d,
                                             b_d, out);
}